// AGNNLayer_4904852652445
// MI455X (gfx1250) — compile-verified
//
#include <hip/hip_runtime.h>

#define DFEAT 64

// ---------- order-preserving float <-> uint encoding for atomic max ----------
__device__ __forceinline__ unsigned enc_f32(float x) {
    unsigned b = __float_as_uint(x);
    return (b & 0x80000000u) ? ~b : (b | 0x80000000u);
}
__device__ __forceinline__ float dec_f32(unsigned e) {
    return __uint_as_float((e & 0x80000000u) ? (e ^ 0x80000000u) : ~e);
}

// ---------- init: zero h accumulator, psum; emax_enc = 0 (== -NaN, below enc(-inf)) ----------
__global__ void agnn_init_kernel(float* __restrict__ h, unsigned* __restrict__ emax,
                                 float* __restrict__ psum, int n_nodes) {
    int i = blockIdx.x * blockDim.x + threadIdx.x;
    if (i < n_nodes) { emax[i] = 0u; psum[i] = 0.0f; }
    if (i < n_nodes * DFEAT) h[i] = 0.0f;
}

// ---------- L2 normalize: one wave32 per node, float2 per lane ----------
__global__ void agnn_normalize_kernel(const float* __restrict__ feat,
                                      float* __restrict__ nh, int n_nodes) {
    int node = (blockIdx.x * blockDim.x + threadIdx.x) >> 5;
    int lane = threadIdx.x & 31;
    if (node >= n_nodes) return;
    const float2 f = *(const float2*)(feat + (long)node * DFEAT + lane * 2);
    float s = f.x * f.x + f.y * f.y;
    #pragma unroll
    for (int o = 16; o > 0; o >>= 1) s += __shfl_xor(s, o, 32);
    float inv = 1.0f / fmaxf(sqrtf(s), 1e-12f);
    float2 r; r.x = f.x * inv; r.y = f.y * inv;
    *(float2*)(nh + (long)node * DFEAT + lane * 2) = r;
}

// ---------- per-edge cosine score + segment max (wave per edge) ----------
__global__ void agnn_edge_score_kernel(const float* __restrict__ nh,
                                       const int* __restrict__ src,
                                       const int* __restrict__ dst,
                                       const float* __restrict__ beta,
                                       float* __restrict__ escore,
                                       unsigned* __restrict__ emax, int n_edges) {
    int e = (blockIdx.x * blockDim.x + threadIdx.x) >> 5;
    int lane = threadIdx.x & 31;
    if (e >= n_edges) return;
    int s = src[e], d = dst[e];
    float2 a = *(const float2*)(nh + (long)s * DFEAT + lane * 2);
    float2 b = *(const float2*)(nh + (long)d * DFEAT + lane * 2);
    float acc = a.x * b.x + a.y * b.y;
    #pragma unroll
    for (int o = 16; o > 0; o >>= 1) acc += __shfl_xor(acc, o, 32);
    if (lane == 0) {
        float sc = beta[0] * acc;
        escore[e] = sc;
        atomicMax(emax + d, enc_f32(sc));   // global_atomic_max_u32
    }
}

// ---------- p = exp(e - emax[dst]); psum[dst] += p (thread per edge) ----------
__global__ void agnn_softmax_kernel(const float* __restrict__ escore,
                                    const int* __restrict__ dst,
                                    const unsigned* __restrict__ emax,
                                    float* __restrict__ p,
                                    float* __restrict__ psum, int n_edges) {
    int e = blockIdx.x * blockDim.x + threadIdx.x;
    if (e >= n_edges) return;
    int d = dst[e];
    float pv = __expf(escore[e] - dec_f32(emax[d]));
    p[e] = pv;
    unsafeAtomicAdd(psum + d, pv);          // global_atomic_add_f32 (no return)
}

// ---------- weighted scatter-sum: h[dst] += (p/psum) * feat[src] (wave per edge) ----------
__global__ void agnn_aggregate_kernel(const float* __restrict__ feat,
                                      const int* __restrict__ src,
                                      const int* __restrict__ dst,
                                      const float* __restrict__ p,
                                      const float* __restrict__ psum,
                                      float* __restrict__ h, int n_edges) {
    int e = (blockIdx.x * blockDim.x + threadIdx.x) >> 5;
    int lane = threadIdx.x & 31;
    if (e >= n_edges) return;
    int s = src[e], d = dst[e];
    float w = p[e] / psum[d];
    float2 f = *(const float2*)(feat + (long)s * DFEAT + lane * 2);
    float* hp = h + (long)d * DFEAT + lane * 2;
    unsafeAtomicAdd(hp,     w * f.x);
    unsafeAtomicAdd(hp + 1, w * f.y);
}

// ---------- projection out = h @ W^T via V_WMMA_F32_16X16X4_F32 ----------
typedef __attribute__((ext_vector_type(2))) float v2f;
typedef __attribute__((ext_vector_type(8))) float v8f;

__global__ void __launch_bounds__(128)
agnn_project_wmma_kernel(const float* __restrict__ h, const float* __restrict__ W,
                         float* __restrict__ out, int n_nodes) {
    const int tile = blockIdx.x;          // 16 nodes per tile
    const int wave = threadIdx.x >> 5;    // 4 waves -> 4 column tiles of 16
    const int lane = threadIdx.x & 31;
    const int m    = lane & 15;           // row within half-wave
    const int hi   = lane >> 4;           // 0: K=k0,k0+1 ; 1: K=k0+2,k0+3
    const int jbase = wave * 16;

    int rowIdx = tile * 16 + m;
    int rowClamped = rowIdx < n_nodes ? rowIdx : n_nodes - 1;  // keep EXEC full for WMMA
    const long rowA = (long)rowClamped * DFEAT;
    const long rowB = (long)(jbase + m) * DFEAT;   // B[k][n] = W[jbase+n][k]

    v8f acc = {};
    #pragma unroll
    for (int k0 = 0; k0 < DFEAT; k0 += 4) {
        int ka = k0 + hi * 2;
        v2f a; a.x = h[rowA + ka]; a.y = h[rowA + ka + 1];
        v2f b; b.x = W[rowB + ka]; b.y = W[rowB + ka + 1];
        // 8 args: (neg_a, A, neg_b, B, c_mod, C, reuse_a, reuse_b)
        acc = __builtin_amdgcn_wmma_f32_16x16x4_f32(
            false, a, false, b, (short)0, acc, false, false);
    }
    // C/D layout: VGPR v -> row (v + hi*8), col = m
    const long col = jbase + m;
    #pragma unroll
    for (int v = 0; v < 8; ++v) {
        int row = tile * 16 + v + hi * 8;
        if (row < n_nodes) out[(long)row * DFEAT + col] = acc[v];
    }
}

extern "C" void kernel_launch(void* const* d_in, const int* in_sizes, int n_in,
                              void* d_out, int out_size, void* d_ws, size_t ws_size,
                              hipStream_t stream) {
    const float* feat = (const float*)d_in[0];
    const float* W    = (const float*)d_in[1];
    const float* beta = (const float*)d_in[2];
    const int*   src  = (const int*)d_in[3];
    const int*   dst  = (const int*)d_in[4];
    const int n_nodes = in_sizes[0] / DFEAT;
    const int n_edges = in_sizes[3];
    float* out = (float*)d_out;

    // carve workspace (256B aligned segments)
    char* ws = (char*)d_ws;
    size_t off = 0;
    auto carve = [&](size_t bytes) -> void* {
        void* ptr = ws + off;
        off = (off + bytes + 255) & ~(size_t)255;
        return ptr;
    };
    float*    nh     = (float*)   carve((size_t)n_nodes * DFEAT * sizeof(float));
    float*    h      = (float*)   carve((size_t)n_nodes * DFEAT * sizeof(float));
    float*    escore = (float*)   carve((size_t)n_edges * sizeof(float));
    float*    p      = (float*)   carve((size_t)n_edges * sizeof(float));
    unsigned* emax   = (unsigned*)carve((size_t)n_nodes * sizeof(unsigned));
    float*    psum   = (float*)   carve((size_t)n_nodes * sizeof(float));

    const int T = 256;
    // 1) init accumulators
    agnn_init_kernel<<<(n_nodes * DFEAT + T - 1) / T, T, 0, stream>>>(h, emax, psum, n_nodes);
    // 2) normalize (wave per node -> 8 nodes per block)
    agnn_normalize_kernel<<<(n_nodes + 7) / 8, T, 0, stream>>>(feat, nh, n_nodes);
    // 3) edge scores + segment max (wave per edge)
    agnn_edge_score_kernel<<<(n_edges + 7) / 8, T, 0, stream>>>(nh, src, dst, beta, escore, emax, n_edges);
    // 4) softmax numerator + denominator
    agnn_softmax_kernel<<<(n_edges + T - 1) / T, T, 0, stream>>>(escore, dst, emax, p, psum, n_edges);
    // 5) weighted scatter-sum (wave per edge)
    agnn_aggregate_kernel<<<(n_edges + 7) / 8, T, 0, stream>>>(feat, src, dst, p, psum, h, n_edges);
    // 6) WMMA projection: one block per 16-node tile, 4 waves cover 64 output cols
    agnn_project_wmma_kernel<<<(n_nodes + 15) / 16, 128, 0, stream>>>(h, W, out, n_nodes);
}